// PtModule_73667279061083
// MI455X (gfx1250) — compile-verified
//
#include <hip/hip_runtime.h>

// y = (x+2) + 3x - (x-1)(x/2) == fma(x, fma(-0.5, x, 4.5), 2.0)
// 8192x8192 fp32 -> 512 MB total traffic; pure HBM-bandwidth-bound (~22us at
// 23.3 TB/s). Optimal shape: unconditional b128 non-temporal loads/stores,
// 32-bit offsets (saddr+voffset addressing), no grid-stride, no predication
// on the interior path, 8 b128 loads in flight per thread.

typedef float f32x4 __attribute__((ext_vector_type(4)));

__device__ __forceinline__ float poly(float x) {
    return __builtin_fmaf(x, __builtin_fmaf(-0.5f, x, 4.5f), 2.0f);
}

__device__ __forceinline__ f32x4 poly4(f32x4 v) {
    f32x4 r;
    r.x = poly(v.x);
    r.y = poly(v.y);
    r.z = poly(v.z);
    r.w = poly(v.w);
    return r;
}

#define TPB 256        // 8 wave32s per block
#define VPT 8          // float4s per thread -> 8 outstanding b128 loads (128 B/thread)
#define TILE (TPB * VPT)

__global__ __launch_bounds__(TPB) void fused_poly_kernel(
    const f32x4* __restrict__ x, f32x4* __restrict__ y, unsigned int n4,
    unsigned int n_rem) {

    // One tile per block; 32-bit element indices (byte offsets < 2^32).
    unsigned int base = blockIdx.x * (unsigned int)TILE;
    unsigned int i0   = base + threadIdx.x;

    if (base + TILE <= n4) {
        // -------- Fast path: full tile, zero predication --------
        f32x4 v[VPT];
#pragma unroll
        for (int k = 0; k < VPT; ++k)
            v[k] = __builtin_nontemporal_load(&x[i0 + k * TPB]);
#pragma unroll
        for (int k = 0; k < VPT; ++k) {
            f32x4 r = poly4(v[k]);
            __builtin_nontemporal_store(r, &y[i0 + k * TPB]);
        }
    } else {
        // -------- Slow path: only the last partial tile ever lands here -----
#pragma unroll
        for (int k = 0; k < VPT; ++k) {
            unsigned int i = i0 + k * TPB;
            if (i < n4) {
                f32x4 r = poly4(__builtin_nontemporal_load(&x[i]));
                __builtin_nontemporal_store(r, &y[i]);
            }
        }
        // Scalar tail (n % 4 != 0). Empty for 8192*8192.
        unsigned int t = threadIdx.x;
        if (t < n_rem) {
            const float* xs = (const float*)x;
            float* ys = (float*)y;
            unsigned long long i = (unsigned long long)n4 * 4ull + t;
            ys[i] = poly(xs[i]);
        }
    }
}

extern "C" void kernel_launch(void* const* d_in, const int* in_sizes, int n_in,
                              void* d_out, int out_size, void* d_ws, size_t ws_size,
                              hipStream_t stream) {
    const float* x = (const float*)d_in[0];
    float* y = (float*)d_out;
    unsigned long long n = (unsigned long long)in_sizes[0];  // 67,108,864

    unsigned int n4 = (unsigned int)(n / 4ull);
    unsigned int n_rem = (unsigned int)(n % 4ull);

    unsigned int tiles = (n4 + TILE - 1) / TILE;   // 8192 blocks for 64Mi elems
    if (tiles == 0) tiles = 1;

    fused_poly_kernel<<<tiles, TPB, 0, stream>>>(
        (const f32x4*)x, (f32x4*)y, n4, n_rem);
}